// EntityAggrNet_81595788689991
// MI455X (gfx1250) — compile-verified
//
#include <hip/hip_runtime.h>
#include <hip/hip_bf16.h>
#include <math.h>

// Problem constants (match the reference)
#define NN 16384      // B*S nodes
#define DD 256        // feature dim
#define EE 524288     // edges
#define ED 64         // embedding depth dim
#define BN_EPS 1e-5f

typedef __attribute__((ext_vector_type(2))) float v2f;
typedef __attribute__((ext_vector_type(8))) float v8f;

// ---------------------------------------------------------------- utilities
__global__ __launch_bounds__(256) void k_zero(float* __restrict__ p, int n) {
    int i = blockIdx.x * blockDim.x + threadIdx.x;
    if (i < n) p[i] = 0.0f;
}

// fast 1/x for positive integer-valued floats: v_rcp_f32 + one Newton step
__device__ __forceinline__ float fast_rcp(float c) {
    float r = __builtin_amdgcn_rcpf(c);
    return r * (2.0f - c * r);
}

// ------------------------------------------------- one-time edge precompute
// wave per edge: cnt[src] += 1 ; aggE[src, 0:64] += emb_table[feat]
__global__ __launch_bounds__(256) void k_edge_pre(const int* __restrict__ edge,
                                                  const int* __restrict__ feat,
                                                  const float* __restrict__ emb,
                                                  float* __restrict__ cnt,
                                                  float* __restrict__ aggE) {
    int wave = (blockIdx.x * blockDim.x + threadIdx.x) >> 5;
    int lane = threadIdx.x & 31;
    if (wave >= EE) return;
    int src = edge[wave];          // edge[0][e]
    int f   = feat[wave];
    if (lane == 0) atomicAdd(&cnt[src], 1.0f);
    const float* er = emb + (size_t)f * ED;
    float* ar = aggE + (size_t)src * ED;
#pragma unroll
    for (int i = 0; i < ED / 32; ++i) {
        int c = lane + 32 * i;
        atomicAdd(&ar[c], er[c]);
    }
}

// ------------------------------------------------- per-layer x scatter-add
// wave per edge: aggX[src, :] += x[dst, :]
__global__ __launch_bounds__(256) void k_scatter_x(const int* __restrict__ edge,
                                                   const float* __restrict__ x,
                                                   float* __restrict__ aggX) {
    int wave = (blockIdx.x * blockDim.x + threadIdx.x) >> 5;
    int lane = threadIdx.x & 31;
    if (wave >= EE) return;
    int src = edge[wave];
    int dst = edge[EE + wave];
    const float* xr = x + (size_t)dst * DD;
    float* ar = aggX + (size_t)src * DD;
#pragma unroll
    for (int i = 0; i < DD / 32; ++i) {
        int c = lane + 32 * i;
        atomicAdd(&ar[c], xr[c]);
    }
}

// ---------------------------------------------------------- WMMA f32 GEMM
// out = (aggX*inv)@Wm + (aggE*inv)@We + x@Ws + bs + 1{cnt>0}*(bm+be)
//
// Block = 256 threads = 8 waves. All 8 waves share one N-tile (nt); the
// K x 16 weight slices of Ws/Wm/We are staged in LDS once per block and
// served to the WMMA B operand via ds_load. Each wave computes one 16x16
// tile with V_WMMA_F32_16X16X4_F32 (full fp32 — compute is nowhere near the
// bottleneck, so no reason to downconvert).
//
// A layout (16x4 f32): lane m = lane&15, half = lane>>4; a = {A[m][k0+2h], A[m][k0+2h+1]}
// B layout (4x16 f32): lane n = lane&15;                 b = {B[k0+2h][n], B[k0+2h+1][n]}
// C/D layout: VGPR r -> row (r + 8*half), col = lane&15
__global__ __launch_bounds__(256) void k_gemm(const float* __restrict__ X,
                                              const float* __restrict__ AggX,
                                              const float* __restrict__ AggE,
                                              const float* __restrict__ cnt,
                                              const float* __restrict__ Wm,
                                              const float* __restrict__ Ws,
                                              const float* __restrict__ We,
                                              const float* __restrict__ bm,
                                              const float* __restrict__ bs,
                                              const float* __restrict__ be,
                                              float* __restrict__ Out) {
    __shared__ float sWs[DD * 16];   // [K=256][16]
    __shared__ float sWm[DD * 16];   // [K=256][16]
    __shared__ float sWe[ED * 16];   // [K=64 ][16]

    int t   = threadIdx.x;
    int nt  = blockIdx.x & 15;        // 16 N-tiles (D/16)
    int mt0 = (blockIdx.x >> 4) * 8;  // 8 M-tiles per block

    // cooperative staging of the three weight K-slices for this N-tile
    for (int i = t; i < DD * 16; i += 256) {
        int k = i >> 4, n = i & 15;
        sWs[i] = Ws[(size_t)k * DD + nt * 16 + n];
        sWm[i] = Wm[(size_t)k * DD + nt * 16 + n];
    }
    for (int i = t; i < ED * 16; i += 256) {
        int k = i >> 4, n = i & 15;
        sWe[i] = We[(size_t)k * DD + nt * 16 + n];
    }
    __syncthreads();

    int wv   = t >> 5;
    int lane = t & 31;
    int half = lane >> 4;
    int m    = lane & 15;
    int mt   = mt0 + wv;
    int row  = mt * 16 + m;               // A-operand row for this lane
    float inv = fast_rcp(fmaxf(cnt[row], 1.0f));

    v8f acc = {};

    // self path: X @ Ws   (K = 256)
    for (int k0 = 0; k0 < DD; k0 += 4) {
        int ka = k0 + 2 * half;
        v2f a = { X[(size_t)row * DD + ka], X[(size_t)row * DD + ka + 1] };
        v2f b = { sWs[ka * 16 + m], sWs[(ka + 1) * 16 + m] };
        acc = __builtin_amdgcn_wmma_f32_16x16x4_f32(false, a, false, b,
                                                    (short)0, acc, false, false);
    }
    // message path: (AggX * inv) @ Wm   (K = 256)
    for (int k0 = 0; k0 < DD; k0 += 4) {
        int ka = k0 + 2 * half;
        v2f a = { AggX[(size_t)row * DD + ka] * inv,
                  AggX[(size_t)row * DD + ka + 1] * inv };
        v2f b = { sWm[ka * 16 + m], sWm[(ka + 1) * 16 + m] };
        acc = __builtin_amdgcn_wmma_f32_16x16x4_f32(false, a, false, b,
                                                    (short)0, acc, false, false);
    }
    // edge path: (AggE * inv) @ We   (K = 64)
    for (int k0 = 0; k0 < ED; k0 += 4) {
        int ka = k0 + 2 * half;
        v2f a = { AggE[(size_t)row * ED + ka] * inv,
                  AggE[(size_t)row * ED + ka + 1] * inv };
        v2f b = { sWe[ka * 16 + m], sWe[(ka + 1) * 16 + m] };
        acc = __builtin_amdgcn_wmma_f32_16x16x4_f32(false, a, false, b,
                                                    (short)0, acc, false, false);
    }

    // epilogue: biases (mean-of-constant bias only contributes where cnt>0)
#pragma unroll
    for (int r = 0; r < 8; ++r) {
        int orow = mt * 16 + r + 8 * half;
        int ocol = nt * 16 + m;
        float ind = (cnt[orow] > 0.0f) ? 1.0f : 0.0f;
        Out[(size_t)orow * DD + ocol] =
            acc[r] + bs[ocol] + ind * (bm[ocol] + be[ocol]);
    }
}

// ----------------------------------------------------- BN column statistics
// block = 256 threads (one per column), each block covers 64 rows
__global__ __launch_bounds__(256) void k_colstats(const float* __restrict__ out,
                                                  float* __restrict__ csum,
                                                  float* __restrict__ csq) {
    int col = threadIdx.x;
    int r0 = blockIdx.x * 64;
    float s = 0.0f, q = 0.0f;
#pragma unroll 4
    for (int r = 0; r < 64; ++r) {
        float v = out[(size_t)(r0 + r) * DD + col];
        s += v;
        q += v * v;
    }
    atomicAdd(&csum[col], s);
    atomicAdd(&csq[col], q);
}

// ------------------------------------------------------- BN apply + ReLU
__global__ __launch_bounds__(256) void k_bn_relu(const float* __restrict__ out,
                                                 const float* __restrict__ csum,
                                                 const float* __restrict__ csq,
                                                 const float* __restrict__ gamma,
                                                 const float* __restrict__ beta,
                                                 float* __restrict__ y) {
    int i = blockIdx.x * blockDim.x + threadIdx.x;
    if (i >= NN * DD) return;
    int col = i & (DD - 1);
    const float rn = 1.0f / (float)NN;
    float mu  = csum[col] * rn;
    float var = csq[col] * rn - mu * mu;
    float v = (out[i] - mu) * rsqrtf(var + BN_EPS) * gamma[col] + beta[col];
    y[i] = fmaxf(v, 0.0f);
}

// ---------------------------------------------------------------- launcher
extern "C" void kernel_launch(void* const* d_in, const int* in_sizes, int n_in,
                              void* d_out, int out_size, void* d_ws, size_t ws_size,
                              hipStream_t stream) {
    const float* data = (const float*)d_in[0];   // [N, 256]
    const float* emb  = (const float*)d_in[1];   // [64, 64]
    const float* Wm   = (const float*)d_in[2];   // [2, 256, 256]
    const float* bm   = (const float*)d_in[3];   // [2, 256]
    const float* Ws   = (const float*)d_in[4];   // [2, 256, 256]
    const float* bs   = (const float*)d_in[5];   // [2, 256]
    const float* We   = (const float*)d_in[6];   // [2, 64, 256]
    const float* be   = (const float*)d_in[7];   // [2, 256]
    const float* gam  = (const float*)d_in[8];   // [2, 256]
    const float* bet  = (const float*)d_in[9];   // [2, 256]
    const int*   edge = (const int*)d_in[10];    // [2, E]
    const int*   feat = (const int*)d_in[11];    // [E]

    // workspace layout (floats)
    float* ws   = (float*)d_ws;
    float* cnt  = ws;                          size_t off = NN;
    float* aggE = ws + off;                    off += (size_t)NN * ED;
    float* aggX = ws + off;                    off += (size_t)NN * DD;
    float* pre  = ws + off;                    off += (size_t)NN * DD;
    float* x1   = ws + off;                    off += (size_t)NN * DD;
    float* csum = ws + off;                    off += DD;
    float* csq  = ws + off;                    off += DD;

    const int ZB = 256;
    // ---- one-time: cnt + embedding aggregation (cnt and aggE are contiguous)
    {
        int n = NN + NN * ED;
        k_zero<<<(n + ZB - 1) / ZB, ZB, 0, stream>>>(cnt, n);
        k_edge_pre<<<(EE * 32) / 256, 256, 0, stream>>>(edge, feat, emb, cnt, aggE);
    }

    const float* x_in = data;
    for (int l = 0; l < 2; ++l) {
        float* x_out = (l == 0) ? x1 : (float*)d_out;

        // zero aggX and column stats (csum/csq contiguous)
        k_zero<<<((NN * DD) + ZB - 1) / ZB, ZB, 0, stream>>>(aggX, NN * DD);
        k_zero<<<(2 * DD + ZB - 1) / ZB, ZB, 0, stream>>>(csum, 2 * DD);

        // scatter x[dst] into aggX[src]
        k_scatter_x<<<(EE * 32) / 256, 256, 0, stream>>>(edge, x_in, aggX);

        // fused 3-GEMM + bias (WMMA f32), LDS-staged weights:
        // grid = (M-tile groups) x (N-tiles) = (N/16/8) * (D/16)
        int blocks = (NN / 16 / 8) * (DD / 16);
        k_gemm<<<blocks, 256, 0, stream>>>(x_in, aggX, aggE, cnt,
                                           Wm + (size_t)l * DD * DD,
                                           Ws + (size_t)l * DD * DD,
                                           We + (size_t)l * ED * DD,
                                           bm + (size_t)l * DD,
                                           bs + (size_t)l * DD,
                                           be + (size_t)l * DD,
                                           pre);

        // batch-norm statistics + apply + ReLU
        k_colstats<<<NN / 64, 256, 0, stream>>>(pre, csum, csq);
        k_bn_relu<<<(NN * DD) / 256, 256, 0, stream>>>(pre, csum, csq,
                                                       gam + (size_t)l * DD,
                                                       bet + (size_t)l * DD,
                                                       x_out);
        x_in = x_out;
    }
}